// DynGraphSAGE_84782654423298
// MI455X (gfx1250) — compile-verified
//
#include <hip/hip_runtime.h>

// DynGraphSAGE forward for MI455X (gfx1250), fp32 via V_WMMA_F32_16X16X4_F32.
// N=50000, NUM=20000, PAST=3, D=128, K1=25, K2=10 — all tile-friendly.

#define GN    50000
#define GNUM  20000
#define GD    128
#define ALPHA 0.2f
#define L2EPS 1e-12f

typedef __attribute__((ext_vector_type(2))) float v2f;
typedef __attribute__((ext_vector_type(8))) float v8f;

__device__ __forceinline__ const float* src_row(const float* head, const float* tail,
                                                int split, int r) {
  return (r < split ? head : tail) + (size_t)r * GD;
}

// ---------------------------------------------------------------------------
// X[i] = [ self_row(i) | mean_k src_row(idx[i,k]) ]   (one wave per row)
// ---------------------------------------------------------------------------
__global__ __launch_bounds__(256) void gather_concat_kernel(
    const float* __restrict__ head, const float* __restrict__ tail, int split,
    const int* __restrict__ idx, int K, float invK,
    float* __restrict__ X, int N)
{
  int row  = blockIdx.x * 8 + (threadIdx.x >> 5);
  int lane = threadIdx.x & 31;
  if (row >= N) return;

  float4 s = ((const float4*)src_row(head, tail, split, row))[lane];
  float4 acc = make_float4(0.f, 0.f, 0.f, 0.f);
  const int* ip = idx + (size_t)row * K;
  for (int k = 0; k < K; ++k) {
    int j = ip[k];
    float4 v = ((const float4*)src_row(head, tail, split, j))[lane];
    acc.x += v.x; acc.y += v.y; acc.z += v.z; acc.w += v.w;
  }
  acc.x *= invK; acc.y *= invK; acc.z *= invK; acc.w *= invK;

  float4* xp = (float4*)(X + (size_t)row * 256);
  xp[lane]      = s;
  xp[32 + lane] = acc;
}

// ---------------------------------------------------------------------------
// X[i] = [ a[i] | b[i] ]  (rows of 128 each)
// ---------------------------------------------------------------------------
__global__ __launch_bounds__(256) void concat_rows_kernel(
    const float* __restrict__ a, const float* __restrict__ b,
    float* __restrict__ X, int N)
{
  int row  = blockIdx.x * 8 + (threadIdx.x >> 5);
  int lane = threadIdx.x & 31;
  if (row >= N) return;
  float4 va = ((const float4*)(a + (size_t)row * GD))[lane];
  float4 vb = ((const float4*)(b + (size_t)row * GD))[lane];
  float4* xp = (float4*)(X + (size_t)row * 256);
  xp[lane]      = va;
  xp[32 + lane] = vb;
}

// ---------------------------------------------------------------------------
// out = (hist[0] + hist[1] + hist[2]) / 3, elementwise (n float4 elements per snap)
// ---------------------------------------------------------------------------
__global__ __launch_bounds__(256) void mean3_kernel(
    const float* __restrict__ h, float* __restrict__ out, int n4)
{
  const float4* p0 = (const float4*)h;
  const float4* p1 = p0 + (size_t)n4;
  const float4* p2 = p1 + (size_t)n4;
  float4* o = (float4*)out;
  const float t = 1.0f / 3.0f;
  for (int i = blockIdx.x * blockDim.x + threadIdx.x; i < n4;
       i += gridDim.x * blockDim.x) {
    float4 a = p0[i], b = p1[i], c = p2[i], r;
    r.x = (a.x + b.x + c.x) * t;
    r.y = (a.y + b.y + c.y) * t;
    r.z = (a.z + b.z + c.z) * t;
    r.w = (a.w + b.w + c.w) * t;
    o[i] = r;
  }
}

// ---------------------------------------------------------------------------
// C[M,128] = A[M,K] @ W[K,128];  if act_norm: leaky(0.2) then row-l2norm.
// Workgroup: 16 rows x 128 cols. 8 waves, each one 16x16 WMMA f32 tile.
// A tile staged in LDS (stride 260 -> bank-conflict-free), W from global (L2).
// ---------------------------------------------------------------------------
__global__ __launch_bounds__(256) void gemm_wmma_kernel(
    const float* __restrict__ A, const float* __restrict__ W,
    float* __restrict__ C, int M, int K, int act_norm)
{
  constexpr int LDA = 260;
  __shared__ float sA[16 * LDA];
  __shared__ float rowsum[16];

  const int tid  = threadIdx.x;
  const int lane = tid & 31;
  const int wave = tid >> 5;
  const int m0   = blockIdx.x * 16;

  if (tid < 16) rowsum[tid] = 0.0f;

  // Cooperative load of the 16 x K A-tile (K is a multiple of 4).
  const int nvec = (16 * K) >> 2;
  for (int v = tid; v < nvec; v += 256) {
    int f = v << 2;
    int r = f / K;
    int c = f - r * K;
    float4 t = *(const float4*)(A + (size_t)(m0 + r) * K + c);
    *(float4*)(&sA[r * LDA + c]) = t;
  }
  __syncthreads();

  const int half = lane >> 4;      // 0: K pair (k,k+1) / rows 0-7 of C
  const int l16  = lane & 15;      // A row / C column within tile
  const int n0   = wave << 4;      // output column base for this wave

  const float* ap = &sA[l16 * LDA + (half << 1)];
  const float* wp = W + ((half << 1) * GD) + n0 + l16;

  v8f acc = {0.f, 0.f, 0.f, 0.f, 0.f, 0.f, 0.f, 0.f};
#pragma unroll 8
  for (int k = 0; k < K; k += 4) {
    v2f a; a.x = ap[k];        a.y = ap[k + 1];
    v2f b; b.x = wp[k * GD];   b.y = wp[k * GD + GD];
    acc = __builtin_amdgcn_wmma_f32_16x16x4_f32(
        false, a, false, b, (short)0, acc, false, false);
  }

  float vals[8];
#pragma unroll
  for (int r = 0; r < 8; ++r) {
    float v = acc[r];
    if (act_norm) v = (v > 0.f) ? v : (ALPHA * v);
    vals[r] = v;
  }

  const int rbase = half << 3;
  if (act_norm) {
    float ss[8];
#pragma unroll
    for (int r = 0; r < 8; ++r) ss[r] = vals[r] * vals[r];
    // reduce over the 16 columns held by this half-wave
#pragma unroll
    for (int m = 1; m <= 8; m <<= 1) {
#pragma unroll
      for (int r = 0; r < 8; ++r) ss[r] += __shfl_xor(ss[r], m, 32);
    }
    if (l16 == 0) {
#pragma unroll
      for (int r = 0; r < 8; ++r) atomicAdd(&rowsum[rbase + r], ss[r]);
    }
    __syncthreads();
#pragma unroll
    for (int r = 0; r < 8; ++r) {
      float nrm = sqrtf(rowsum[rbase + r]);
      vals[r] *= 1.0f / fmaxf(nrm, L2EPS);
    }
  }

#pragma unroll
  for (int r = 0; r < 8; ++r) {
    int row = rbase + r;                     // C/D layout: VGPR r -> M=r (+8 upper half)
    C[(size_t)(m0 + row) * GD + n0 + l16] = vals[r];
  }
}

// ---------------------------------------------------------------------------
extern "C" void kernel_launch(void* const* d_in, const int* in_sizes, int n_in,
                              void* d_out, int out_size, void* d_ws, size_t ws_size,
                              hipStream_t stream)
{
  const int N = GN, NUM = GNUM, D = GD;
  const float* feats = (const float*)d_in[0];
  const int*   idx1  = (const int*)  d_in[1];
  const int*   idx2  = (const int*)  d_in[2];
  const float* hist1 = (const float*)d_in[3];
  const float* hist2 = (const float*)d_in[4];
  const float* W1    = (const float*)d_in[5];
  const float* W2    = (const float*)d_in[6];
  const float* Whis  = (const float*)d_in[7];
  const float* WT    = (const float*)d_in[8];

  float* out_h1 = (float*)d_out;
  float* out_h2 = out_h1 + (size_t)N * D;
  float* out_ft = out_h2 + (size_t)N * D;

  float* X  = (float*)d_ws;                 // [N, 256]
  float* f1 = X  + (size_t)N * 256;         // [NUM, 128]
  float* Hm = f1 + (size_t)NUM * D;         // [NUM, 128]
  float* TF = Hm + (size_t)NUM * D;         // [NUM, 128]

  dim3 b(256);

  // ---- first-order SAGE ----
  gather_concat_kernel<<<N / 8, b, 0, stream>>>(feats, feats, N, idx1, 25,
                                                1.0f / 25.0f, X, N);
  gemm_wmma_kernel<<<N / 16, b, 0, stream>>>(X, W1, out_h1, N, 256, 1);

  // ---- temporal fusion 1 ----
  mean3_kernel<<<640, b, 0, stream>>>(hist1, Hm, NUM * D / 4);
  gemm_wmma_kernel<<<NUM / 16, b, 0, stream>>>(Hm, Whis, TF, NUM, 128, 0);
  concat_rows_kernel<<<NUM / 8, b, 0, stream>>>(out_h1, TF, X, NUM);
  gemm_wmma_kernel<<<NUM / 16, b, 0, stream>>>(X, WT, f1, NUM, 256, 1);

  // ---- second-order SAGE over h1_copy = [f1 ; h1[NUM:]] ----
  gather_concat_kernel<<<N / 8, b, 0, stream>>>(f1, out_h1, NUM, idx2, 10,
                                                1.0f / 10.0f, X, N);
  gemm_wmma_kernel<<<N / 16, b, 0, stream>>>(X, W2, out_h2, N, 256, 1);

  // ---- temporal fusion 2 ----
  mean3_kernel<<<640, b, 0, stream>>>(hist2, Hm, NUM * D / 4);
  gemm_wmma_kernel<<<NUM / 16, b, 0, stream>>>(Hm, Whis, TF, NUM, 128, 0);
  concat_rows_kernel<<<NUM / 8, b, 0, stream>>>(out_h2, TF, X, NUM);
  gemm_wmma_kernel<<<NUM / 16, b, 0, stream>>>(X, WT, out_ft, NUM, 256, 1);

  // feat tail = h2[NUM:]
  hipMemcpyAsync(out_ft + (size_t)NUM * D, out_h2 + (size_t)NUM * D,
                 (size_t)(N - NUM) * D * sizeof(float),
                 hipMemcpyDeviceToDevice, stream);
}